// ECE_LOSS_equal_width_20512763806003
// MI455X (gfx1250) — compile-verified
//
#include <hip/hip_runtime.h>
#include <hip/hip_bf16.h>

#define ECE_NBINS 15
#define ECE_C     128

// gfx1250 async global->LDS path (ASYNCcnt-tracked), with fallback to direct
// VGPR loads if the toolchain doesn't expose the builtins.
#if __has_builtin(__builtin_amdgcn_global_load_async_to_lds_b128) && \
    __has_builtin(__builtin_amdgcn_s_wait_asynccnt)
#define ECE_ASYNC 1
#else
#define ECE_ASYNC 0
#endif

#define ECE_AS1 __attribute__((address_space(1)))
#define ECE_AS3 __attribute__((address_space(3)))

typedef int ece_v4i __attribute__((vector_size(16)));   // 128-bit payload type

// ---------------------------------------------------------------------------
// Kernel 1: zero the 45-float bin accumulators in workspace.
// ---------------------------------------------------------------------------
__global__ void ece_zero_kernel(float* __restrict__ bins) {
    int i = threadIdx.x;
    if (i < 3 * ECE_NBINS) bins[i] = 0.0f;
}

// ---------------------------------------------------------------------------
// Kernel 2: one wave32 per row (32 lanes x float4 == 128 logits == one row).
// Rows are streamed HBM -> LDS with double-buffered async copies; wave
// reduces max/argmax and sum(exp(x-max)); lane 0 bins into LDS; block
// flushes 45 FP32 global atomics at the end.
// ---------------------------------------------------------------------------
__global__ __launch_bounds__(256) void ece_main_kernel(
        const float* __restrict__ y_pred,
        const int*   __restrict__ y_true,
        float*       __restrict__ bins,
        int nrows) {
    __shared__ float s_cnt[ECE_NBINS];
    __shared__ float s_conf[ECE_NBINS];
    __shared__ float s_acc[ECE_NBINS];
#if ECE_ASYNC
    __shared__ __align__(16) float rowbuf[8][2][ECE_C];   // 8 waves x 2 rows
#endif

    const int tid = threadIdx.x;
    if (tid < ECE_NBINS) {
        s_cnt[tid]  = 0.0f;
        s_conf[tid] = 0.0f;
        s_acc[tid]  = 0.0f;
    }
    __syncthreads();

    const int lane          = tid & 31;
    const int wib           = tid >> 5;
    const int wavesPerBlock = blockDim.x >> 5;
    const int gwave    = blockIdx.x * wavesPerBlock + wib;
    const int numWaves = gridDim.x * wavesPerBlock;

    // Shared per-row epilogue: wave reductions + binning.
    auto process_row = [&](float4 v, int row) {
        // Lane-local max + argmax (first index wins ties, like jnp.argmax).
        float m  = v.x; int mi = lane * 4;
        if (v.y > m) { m = v.y; mi = lane * 4 + 1; }
        if (v.z > m) { m = v.z; mi = lane * 4 + 2; }
        if (v.w > m) { m = v.w; mi = lane * 4 + 3; }
        #pragma unroll
        for (int off = 16; off > 0; off >>= 1) {
            float om = __shfl_xor(m,  off, 32);
            int   oi = __shfl_xor(mi, off, 32);
            if (om > m || (om == m && oi < mi)) { m = om; mi = oi; }
        }
        float s = __expf(v.x - m) + __expf(v.y - m)
                + __expf(v.z - m) + __expf(v.w - m);
        #pragma unroll
        for (int off = 16; off > 0; off >>= 1) {
            s += __shfl_xor(s, off, 32);
        }
        if (lane == 0) {
            float conf = 1.0f / s;                  // max(softmax(row))
            float acc  = (mi == y_true[row]) ? 1.0f : 0.0f;
            int b = (int)ceilf(conf * (float)ECE_NBINS) - 1;
            b = b < 0 ? 0 : (b > ECE_NBINS - 1 ? ECE_NBINS - 1 : b);
            atomicAdd(&s_cnt[b],  1.0f);            // ds_add_f32
            atomicAdd(&s_conf[b], conf);
            atomicAdd(&s_acc[b],  acc);
        }
    };

#if ECE_ASYNC
    float* bufs[2] = { &rowbuf[wib][0][0], &rowbuf[wib][1][0] };

    auto issue_async = [&](int row, float* dst) {
        float* src = const_cast<float*>(y_pred + (size_t)row * ECE_C + lane * 4);
        __builtin_amdgcn_global_load_async_to_lds_b128(
            (ECE_AS1 ece_v4i*)src,                  // global_load_async_to_lds_b128
            (ECE_AS3 ece_v4i*)(dst + lane * 4),
            0, 0);
    };

    if (gwave < nrows) {
        issue_async(gwave, bufs[0]);                // prime the pipeline
        int cur = 0;
        for (int row = gwave; row < nrows; row += numWaves) {
            int nrow = row + numWaves;
            if (nrow < nrows) {
                issue_async(nrow, bufs[cur ^ 1]);   // keep 2 rows in flight
                int pf = nrow + numWaves;           // L2 prefetch 2 rows ahead
                if (pf < nrows) {
                    __builtin_prefetch(y_pred + (size_t)pf * ECE_C + lane * 4, 0, 0);
                }
                __builtin_amdgcn_s_wait_asynccnt(1); // current buffer landed
            } else {
                __builtin_amdgcn_s_wait_asynccnt(0);
            }
            asm volatile("" ::: "memory");
            float4 v = *(reinterpret_cast<const float4*>(bufs[cur]) + lane); // ds_load_b128
            process_row(v, row);
            cur ^= 1;
        }
    }
#else
    for (int row = gwave; row < nrows; row += numWaves) {
        int nrow = row + numWaves;
        if (nrow < nrows) {
            __builtin_prefetch(y_pred + (size_t)nrow * ECE_C + lane * 4, 0, 0);
        }
        float4 v = *(reinterpret_cast<const float4*>(
                         y_pred + (size_t)row * ECE_C) + lane);  // global_load_b128
        process_row(v, row);
    }
#endif

    __syncthreads();
    if (tid < ECE_NBINS) {
        unsafeAtomicAdd(&bins[tid],                 s_cnt[tid]);  // global_atomic_add_f32
        unsafeAtomicAdd(&bins[ECE_NBINS + tid],     s_conf[tid]);
        unsafeAtomicAdd(&bins[2 * ECE_NBINS + tid], s_acc[tid]);
    }
}

// ---------------------------------------------------------------------------
// Kernel 3: fold 15 bins into the scalar ECE.
//   |avg_conf - avg_acc| * cnt/N  ==  |conf_sum - acc_sum| / N  (cnt > 0)
// ---------------------------------------------------------------------------
__global__ void ece_final_kernel(const float* __restrict__ bins,
                                 float* __restrict__ out,
                                 int nrows) {
    if (threadIdx.x == 0 && blockIdx.x == 0) {
        float e = 0.0f;
        #pragma unroll
        for (int b = 0; b < ECE_NBINS; ++b) {
            float cnt = bins[b];
            if (cnt > 0.0f) {
                float avg_conf = bins[ECE_NBINS + b]     / cnt;
                float avg_acc  = bins[2 * ECE_NBINS + b] / cnt;
                e += fabsf(avg_conf - avg_acc) * (cnt / (float)nrows);
            }
        }
        out[0] = e;
    }
}

// ---------------------------------------------------------------------------
extern "C" void kernel_launch(void* const* d_in, const int* in_sizes, int n_in,
                              void* d_out, int out_size, void* d_ws, size_t ws_size,
                              hipStream_t stream) {
    const float* y_pred = (const float*)d_in[0];
    const int*   y_true = (const int*)d_in[1];
    float*       bins   = (float*)d_ws;      // 45 floats of scratch
    float*       out    = (float*)d_out;

    const int nrows = in_sizes[1];           // N (y_true element count)

    ece_zero_kernel<<<1, 64, 0, stream>>>(bins);

    // Persistent grid: 2048 blocks x 8 wave32 = 16K waves, ~64 rows each.
    ece_main_kernel<<<2048, 256, 0, stream>>>(y_pred, y_true, bins, nrows);

    ece_final_kernel<<<1, 32, 0, stream>>>(bins, out, nrows);
}